// GATConv_74904229642448
// MI455X (gfx1250) — compile-verified
//
#include <hip/hip_runtime.h>

// GATConv on MI455X (gfx1250).
//
// Algebraic simplification: the reference normalizes alpha over axis 1 and then
// contracts ONLY alpha over that axis ('bij,bjk->bkj' — i appears solely in
// alpha), so Sum_i alpha[b,i,j] == 1 and the whole dense-attention pipeline
// (adj, leaky-relu, exp, softmax) cancels. Exact result:
//     out[n, k*HEADS + h] = (x @ W^T)[n, h*OUT_C + k] + bias[k*HEADS + h]
// => one fp32 GEMM (4096x512x256) + permuted bias-add, done with
// V_WMMA_F32_16X16X4_F32 to keep full fp32 precision.

typedef __attribute__((ext_vector_type(2))) float v2f;
typedef __attribute__((ext_vector_type(8))) float v8f;

#define N_NODES 4096
#define IN_C    256
#define OUT_C   64
#define HEADS   8
#define OUTF    (HEADS * OUT_C)   // 512

__global__ __launch_bounds__(256)
void gat_wmma_f32_kernel(const float* __restrict__ x,     // [4096, 256]
                         const float* __restrict__ W,     // [512, 256]
                         const float* __restrict__ bias,  // [512]
                         float* __restrict__ out)         // [4096, 512]
{
    const int lane = threadIdx.x & 31;    // wave32 lane
    const int head = threadIdx.x >> 5;    // wave id 0..7 == head / 64-col strip
    const int m0   = blockIdx.x << 4;     // 16 output rows per block
    const int r16  = lane & 15;           // A row-in-tile / B col-in-tile
    const int kq   = (lane >> 4) << 1;    // lanes 0-15 -> K 0,1 ; lanes 16-31 -> K 2,3

    // A fragment source: x[m0 + r16, kq + k .. +1]
    const float* __restrict__ aPtr = x + (size_t)(m0 + r16) * IN_C + kq;
    // B fragment source: B[k][n] = W[n][k]; wave's cols n = head*64 + t*16 + r16
    const float* __restrict__ bPtr[4];
#pragma unroll
    for (int t = 0; t < 4; ++t)
        bPtr[t] = W + (size_t)(head * 64 + t * 16 + r16) * IN_C + kq;

    v8f acc[4];
#pragma unroll
    for (int t = 0; t < 4; ++t)
#pragma unroll
        for (int r = 0; r < 8; ++r)
            acc[t][r] = 0.0f;

#pragma unroll 4
    for (int k = 0; k < IN_C; k += 4) {
        // Prefetch 64 floats (16 k-steps) ahead (speculative; OOB translation failures dropped).
        __builtin_prefetch((const void*)(aPtr + k + 64), 0, 3);
        __builtin_prefetch((const void*)(bPtr[0] + k + 64), 0, 3);

        const v2f a = *(const v2f*)(aPtr + k);
        v2f b[4];
#pragma unroll
        for (int t = 0; t < 4; ++t)
            b[t] = *(const v2f*)(bPtr[t] + k);

#pragma unroll
        for (int t = 0; t < 4; ++t) {
            // D = A(16x4) x B(4x16) + C ; 8 args: neg_a, A, neg_b, B, c_mod, C, reuse_a, reuse_b
            acc[t] = __builtin_amdgcn_wmma_f32_16x16x4_f32(
                false, a, false, b[t], (short)0, acc[t], false, false);
        }
    }

    // Epilogue with the 'bkj' permutation: out[row, kk*8 + head] = acc + bias.
    // C/D layout: VGPR r holds M=r (lanes 0-15) / M=8+r (lanes 16-31), N = lane%16.
    const int rowBase = m0 + ((lane >> 4) << 3);
#pragma unroll
    for (int t = 0; t < 4; ++t) {
        const int kk     = t * 16 + r16;       // channel within head (0..63)
        const int outcol = kk * HEADS + head;  // permuted output column
        const float bv   = bias[outcol];
#pragma unroll
        for (int r = 0; r < 8; ++r) {
            out[(size_t)(rowBase + r) * OUTF + outcol] = acc[t][r] + bv;
        }
    }
}

extern "C" void kernel_launch(void* const* d_in, const int* in_sizes, int n_in,
                              void* d_out, int out_size, void* d_ws, size_t ws_size,
                              hipStream_t stream) {
    // setup_inputs order: adj, x, W, att_src, att_dst, bias
    // adj / att_src / att_dst are algebraically eliminated (softmax sums to 1).
    const float* x    = (const float*)d_in[1];
    const float* W    = (const float*)d_in[2];
    const float* bias = (const float*)d_in[5];
    float* out        = (float*)d_out;

    dim3 grid(N_NODES / 16);   // 256 blocks: one 16-row tile each
    dim3 block(256);           // 8 wave32 waves: one head each
    hipLaunchKernelGGL(gat_wmma_f32_kernel, grid, block, 0, stream,
                       x, W, bias, out);
}